// EuclideanEmbedding_82712480186400
// MI455X (gfx1250) — compile-verified
//
#include <hip/hip_runtime.h>
#include <hip/hip_bf16.h>

typedef float v4f __attribute__((ext_vector_type(4)));

// ---------------------------------------------------------------------------
// Kernel 1: zero the [E,16] output via CDNA5 async store-from-LDS.
// Each lane owns a 16B LDS slot of zeros; the async engine streams it to
// global memory with a non-temporal hint (don't pollute L2 — the scatter
// kernel wants the 6.4MB hot atomic region to stay resident there).
// ---------------------------------------------------------------------------
__global__ __launch_bounds__(256) void ee_zero_async_kernel(float* __restrict__ out,
                                                            long long n) {
    __shared__ __align__(16) float zbuf[256 * 4];
    const int t = threadIdx.x;
    v4f z = {0.0f, 0.0f, 0.0f, 0.0f};
    *(v4f*)(zbuf + t * 4) = z;
    // LDS byte offset of this lane's slot (generic LDS pointer low 32 bits).
    unsigned lds_addr = (unsigned)(unsigned long long)(zbuf + t * 4);
    // Ensure the DS store landed before the async engine reads LDS.
    asm volatile("s_wait_dscnt 0x0" ::: "memory");

    const long long n4 = n >> 2;
    long long i = (long long)blockIdx.x * blockDim.x + t;
    const long long stride = (long long)gridDim.x * blockDim.x;
    for (; i < n4; i += stride) {
        unsigned long long gaddr = (unsigned long long)out + ((unsigned long long)i << 4);
        asm volatile("global_store_async_from_lds_b128 %0, %1, off th:TH_STORE_NT"
                     :: "v"(gaddr), "v"(lds_addr) : "memory");
    }
    // Scalar tail (out_size is 4-divisible here, but stay generic).
    if (blockIdx.x == 0 && t < (int)(n & 3)) out[(n4 << 2) + t] = 0.0f;
    asm volatile("s_wait_asynccnt 0x0" ::: "memory");
}

// ---------------------------------------------------------------------------
// Kernel 2: per-edge cutoff-weighted scatter-add.
// One thread per edge: NT-load 16 sender floats (4 x b128), scale by
// w = 0.5*(cos(pi*r/5)+1)/32 when r < rcut, then 16 fire-and-forget
// global_atomic_add_f32 into out[receiver*16 .. +15] (L2-resident region).
// Edges past the cutoff skip both the 64B load and all atomics.
// ---------------------------------------------------------------------------
#define EE_ATOM(val, ofs)                                                     \
    asm volatile("global_atomic_add_f32 %0, %1, off offset:" #ofs             \
                 :: "v"(base), "v"(val) : "memory")

__global__ __launch_bounds__(256) void ee_scatter_kernel(
        const float* __restrict__ senders,
        const float* __restrict__ lengths,
        const int*   __restrict__ receivers,
        float* __restrict__ out,
        int E) {
    int i = blockIdx.x * blockDim.x + threadIdx.x;
    const int stride = gridDim.x * blockDim.x;
    for (int e = i; e < E; e += stride) {
        float r = __builtin_nontemporal_load(lengths + e);
        if (r < 5.0f) {
            // w = 0.5*(cos(pi*r/5)+1) * (1/32)
            float w = 0.015625f * (__cosf(0.6283185307179586f * r) + 1.0f);
            int rec = __builtin_nontemporal_load(receivers + e);
            const v4f* s4 = (const v4f*)(senders + (size_t)e * 16);
            v4f a = __builtin_nontemporal_load(s4 + 0);
            v4f b = __builtin_nontemporal_load(s4 + 1);
            v4f c = __builtin_nontemporal_load(s4 + 2);
            v4f d = __builtin_nontemporal_load(s4 + 3);
            a *= w; b *= w; c *= w; d *= w;
            unsigned long long base =
                (unsigned long long)(out + (size_t)rec * 16);
            EE_ATOM(a.x, 0);  EE_ATOM(a.y, 4);  EE_ATOM(a.z, 8);  EE_ATOM(a.w, 12);
            EE_ATOM(b.x, 16); EE_ATOM(b.y, 20); EE_ATOM(b.z, 24); EE_ATOM(b.w, 28);
            EE_ATOM(c.x, 32); EE_ATOM(c.y, 36); EE_ATOM(c.z, 40); EE_ATOM(c.w, 44);
            EE_ATOM(d.x, 48); EE_ATOM(d.y, 52); EE_ATOM(d.z, 56); EE_ATOM(d.w, 60);
        }
    }
}

extern "C" void kernel_launch(void* const* d_in, const int* in_sizes, int n_in,
                              void* d_out, int out_size, void* d_ws, size_t ws_size,
                              hipStream_t stream) {
    const float* senders   = (const float*)d_in[0];   // [E,16] f32
    const float* lengths   = (const float*)d_in[1];   // [E,1]  f32
    // d_in[2] = vectors [E,3] — unused by the reference's output
    const int*   receivers = (const int*)d_in[3];     // [E]    int
    float* out = (float*)d_out;                       // [E,16] f32

    const int E = in_sizes[3];
    const long long n = (long long)out_size;

    // Pass 1: zero-fill output via async LDS->global stores (NT).
    {
        long long n4 = (n + 3) >> 2;
        int blocks = (int)((n4 + 255) / 256);
        if (blocks > 8192) blocks = 8192;
        if (blocks < 1) blocks = 1;
        ee_zero_async_kernel<<<blocks, 256, 0, stream>>>(out, n);
    }
    // Pass 2: weighted scatter-add (stream-ordered after the fill).
    {
        int blocks = (E + 255) / 256;
        if (blocks < 1) blocks = 1;
        ee_scatter_kernel<<<blocks, 256, 0, stream>>>(senders, lengths,
                                                      receivers, out, E);
    }
}